// HeteroGNN_15556371546392
// MI455X (gfx1250) — compile-verified
//
#include <hip/hip_runtime.h>
#include <hip/hip_bf16.h>

// ---------------------------------------------------------------------------
// Problem constants (match reference)
// ---------------------------------------------------------------------------
#define NT   4
#define NN   16384
#define ET   14
#define EE   65536
#define CC   64
#define CIN  11
#define EPF  524288
#define SCALE_F (0.125f)   // 1/sqrt(64)

__constant__ int SRC_A[ET] = {0, 0, 0, 1, 1, 2, 1, 2, 3, 2, 3, 1, 2, 3};
__constant__ int DST_A[ET] = {1, 2, 3, 2, 3, 3, 0, 0, 0, 1, 2, 1, 2, 3};

typedef __attribute__((ext_vector_type(2))) float v2f;
typedef __attribute__((ext_vector_type(8))) float v8f;

// Branch-free guarded load: clamp index in-bounds, select 0 for padding.
// For KPAD==CREAL the guard folds away entirely at compile time.
template<int KPAD, int CREAL>
__device__ __forceinline__ float ldGuard(const float* __restrict__ p, int k, int stride0)
{
    if constexpr (KPAD == CREAL) {
        return p[(size_t)k * stride0];
    } else {
        const int kc = (k < CREAL) ? k : (CREAL - 1);
        const float v = p[(size_t)kc * stride0];
        return (k < CREAL) ? v : 0.0f;
    }
}

// ---------------------------------------------------------------------------
// Per-edge-type projection GEMM:  Y[t] = X[sel[t]] @ W[t] + b[t]
// X: (NT, NN, CREAL) f32, W: (ET, CREAL, CC), b: (ET, CC), Y: (ET, NN, CC)
// One wave32 computes a 16x64 output strip (4 col tiles, 4 accumulators)
// via V_WMMA_F32_16X16X4_F32; K fully unrolled at compile time.
// ---------------------------------------------------------------------------
template<int KPAD, int CREAL>
__global__ void proj_gemm_wmma(const float* __restrict__ X,
                               const float* __restrict__ W,
                               const float* __restrict__ bias,
                               float* __restrict__ Y,
                               int useSrc)
{
    const int lane   = threadIdx.x & 31;
    const int waveId = threadIdx.x >> 5;
    const int tile   = blockIdx.x * 8 + waveId;      // 8 waves / block
    const int TILES  = ET * (NN / 16);               // 14336 row-strips
    if (tile >= TILES) return;

    const int t    = tile >> 10;                     // / (NN/16)
    const int row0 = (tile & 1023) * 16;

    const int sel = useSrc ? SRC_A[t] : DST_A[t];
    const float* __restrict__ Xb = X + (size_t)sel * NN * CREAL;
    const float* __restrict__ Wb = W + (size_t)t * CREAL * CC;
    const float* __restrict__ bb = bias + (size_t)t * CC;
    float* __restrict__ Yb       = Y + ((size_t)t * NN) * CC;

    const int half = lane >> 4;                      // 0 or 1
    const int l16  = lane & 15;
    const int m    = l16;                            // A row (M) for this lane
    const int n    = l16;                            // B/D col (N) for this lane

    const float* __restrict__ Arow = Xb + (size_t)(row0 + m) * CREAL;

    v8f acc0 = {}, acc1 = {}, acc2 = {}, acc3 = {};

#pragma unroll
    for (int k0 = 0; k0 < KPAD; k0 += 4) {
        const int ka = k0 + 2 * half;                // this lane's two K slots
        v2f av;
        if constexpr (KPAD == CREAL && (CREAL % 2) == 0) {
            av = *(const v2f*)(Arow + ka);           // global_load_b64
        } else {
            av[0] = ldGuard<KPAD, CREAL>(Arow, ka + 0, 1);
            av[1] = ldGuard<KPAD, CREAL>(Arow, ka + 1, 1);
        }
        const float* __restrict__ Wk0 = Wb + n;      // + k*CC per element
#pragma unroll
        for (int ct = 0; ct < 4; ++ct) {
            v2f bv;
            bv[0] = ldGuard<KPAD, CREAL>(Wk0 + ct * 16, ka + 0, CC);
            bv[1] = ldGuard<KPAD, CREAL>(Wk0 + ct * 16, ka + 1, CC);
            switch (ct) {
            case 0: acc0 = __builtin_amdgcn_wmma_f32_16x16x4_f32(false, av, false, bv, (short)0, acc0, false, false); break;
            case 1: acc1 = __builtin_amdgcn_wmma_f32_16x16x4_f32(false, av, false, bv, (short)0, acc1, false, false); break;
            case 2: acc2 = __builtin_amdgcn_wmma_f32_16x16x4_f32(false, av, false, bv, (short)0, acc2, false, false); break;
            case 3: acc3 = __builtin_amdgcn_wmma_f32_16x16x4_f32(false, av, false, bv, (short)0, acc3, false, false); break;
            }
        }
    }

#pragma unroll
    for (int ct = 0; ct < 4; ++ct) {
        const v8f& acc = (ct == 0) ? acc0 : (ct == 1) ? acc1 : (ct == 2) ? acc2 : acc3;
        const float bval = bb[ct * 16 + n];
#pragma unroll
        for (int r = 0; r < 8; ++r) {
            const int mOut = r + 8 * half;           // D layout: VGPR r -> M=r / M=r+8
            Yb[(size_t)(row0 + mOut) * CC + ct * 16 + n] = acc[r] + bval;
        }
    }
}

// ---------------------------------------------------------------------------
// Init segment-max (-inf) and segment-sum (0) buffers
// ---------------------------------------------------------------------------
__global__ void init_seg(float* __restrict__ M, float* __restrict__ S, int n)
{
    int i = blockIdx.x * blockDim.x + threadIdx.x;
    if (i < n) { M[i] = -INFINITY; S[i] = 0.0f; }
}

__device__ __forceinline__ float waveReduceSum(float v)
{
    for (int off = 16; off > 0; off >>= 1)
        v += __shfl_xor(v, off, 32);
    return v;
}

__device__ __forceinline__ void atomicMaxFloat(float* addr, float val)
{
    if (val >= 0.0f) atomicMax((int*)addr, __float_as_int(val));
    else             atomicMin((unsigned int*)addr, __float_as_uint(val));
}

// ---------------------------------------------------------------------------
// Attention pass 1: logit = dot(q[dst], k[src] + ee) * SCALE ; segment max
// One wave per edge; lanes cover C=64 (2 each).
// ---------------------------------------------------------------------------
__global__ void attn_logit(const float* __restrict__ Q, const float* __restrict__ K,
                           const float* __restrict__ ea, const float* __restrict__ We,
                           const int* __restrict__ ei,
                           float* __restrict__ LOGIT, float* __restrict__ M)
{
    const int lane = threadIdx.x & 31;
    const int g    = blockIdx.x * 8 + (threadIdx.x >> 5);
    if (g >= ET * EE) return;
    const int t = g >> 16, e = g & (EE - 1);

    const int src = ei[(size_t)(t * 2 + 0) * EE + e];
    const int dst = ei[(size_t)(t * 2 + 1) * EE + e];
    const float ea0 = ea[((size_t)t * EE + e) * 2 + 0];
    const float ea1 = ea[((size_t)t * EE + e) * 2 + 1];

    const float* qr  = Q  + ((size_t)t * NN + dst) * CC;
    const float* kr  = K  + ((size_t)t * NN + src) * CC;
    const float* we0 = We + (size_t)t * 2 * CC;
    const float* we1 = we0 + CC;

    float p = 0.0f;
#pragma unroll
    for (int h = 0; h < 2; ++h) {
        const int c = lane + 32 * h;
        const float ke = kr[c] + ea0 * we0[c] + ea1 * we1[c];
        p += qr[c] * ke;
    }
    p = waveReduceSum(p) * SCALE_F;
    if (lane == 0) {
        LOGIT[g] = p;
        atomicMaxFloat(&M[(size_t)t * NN + dst], p);
    }
}

// ---------------------------------------------------------------------------
// Attention pass 2: a = exp(logit - m[dst]); segment sum. Thread per edge.
// ---------------------------------------------------------------------------
__global__ void attn_expsum(const int* __restrict__ ei,
                            float* __restrict__ LOGIT,
                            const float* __restrict__ M, float* __restrict__ S)
{
    const int g = blockIdx.x * blockDim.x + threadIdx.x;
    if (g >= ET * EE) return;
    const int t = g >> 16, e = g & (EE - 1);
    const int dst = ei[(size_t)(t * 2 + 1) * EE + e];
    const float a = expf(LOGIT[g] - M[(size_t)t * NN + dst]);
    LOGIT[g] = a;
    atomicAdd(&S[(size_t)t * NN + dst], a);
}

// ---------------------------------------------------------------------------
// Attention pass 3: MSG[dst] += alpha * (v[src] + ee).  Wave per edge.
// ---------------------------------------------------------------------------
__global__ void attn_scatter(const float* __restrict__ V, const float* __restrict__ ea,
                             const float* __restrict__ We, const int* __restrict__ ei,
                             const float* __restrict__ LOGIT, const float* __restrict__ S,
                             float* __restrict__ MSG)
{
    const int lane = threadIdx.x & 31;
    const int g    = blockIdx.x * 8 + (threadIdx.x >> 5);
    if (g >= ET * EE) return;
    const int t = g >> 16, e = g & (EE - 1);

    const int src = ei[(size_t)(t * 2 + 0) * EE + e];
    const int dst = ei[(size_t)(t * 2 + 1) * EE + e];
    const float ea0 = ea[((size_t)t * EE + e) * 2 + 0];
    const float ea1 = ea[((size_t)t * EE + e) * 2 + 1];
    const float alpha = LOGIT[g] / (S[(size_t)t * NN + dst] + 1e-16f);

    const float* vr  = V  + ((size_t)t * NN + src) * CC;
    const float* we0 = We + (size_t)t * 2 * CC;
    const float* we1 = we0 + CC;
    float* mr = MSG + ((size_t)t * NN + dst) * CC;

#pragma unroll
    for (int h = 0; h < 2; ++h) {
        const int c = lane + 32 * h;
        const float ve = vr[c] + ea0 * we0[c] + ea1 * we1[c];
        atomicAdd(&mr[c], alpha * ve);
    }
}

// ---------------------------------------------------------------------------
// H[nt][n][c] = relu( sum over {t : DST[t]==nt} MSG[t][n][c] )
// ---------------------------------------------------------------------------
__global__ void reduce_relu(const float* __restrict__ MSG, float* __restrict__ H)
{
    const int idx = blockIdx.x * blockDim.x + threadIdx.x;
    if (idx >= NT * NN * CC) return;
    const int nt = idx / (NN * CC);
    const int nc = idx % (NN * CC);
    float s = 0.0f;
#pragma unroll
    for (int t = 0; t < ET; ++t)
        if (DST_A[t] == nt) s += MSG[(size_t)t * NN * CC + nc];
    H[idx] = fmaxf(s, 0.0f);
}

// ---------------------------------------------------------------------------
// Final linear (C=64 -> 4) + write cols 0,1, zero cols 2,3, stash |V| and theta
// ---------------------------------------------------------------------------
__global__ void final_linear(const float* __restrict__ H, const float* __restrict__ linW,
                             float* __restrict__ out, float* __restrict__ Vm,
                             float* __restrict__ Th)
{
    const int u = blockIdx.x * blockDim.x + threadIdx.x;
    if (u >= NT * NN) return;
    const float* hr = H + (size_t)u * CC;
    float x0 = 0.f, x1 = 0.f;
#pragma unroll 8
    for (int c = 0; c < CC; ++c) {
        const float h = hr[c];
        x0 += h * linW[c * 4 + 0];
        x1 += h * linW[c * 4 + 1];
    }
    out[(size_t)u * 4 + 0] = x0;
    out[(size_t)u * 4 + 1] = x1;
    out[(size_t)u * 4 + 2] = 0.0f;
    out[(size_t)u * 4 + 3] = 0.0f;
    Vm[u] = fabsf(x0);
    Th[u] = x1;
}

// ---------------------------------------------------------------------------
// Power-flow scatter: P,Q segment sums over pf_i
// ---------------------------------------------------------------------------
__global__ void powerflow(const float* __restrict__ pf_attr,
                          const int* __restrict__ pf_i, const int* __restrict__ pf_j,
                          const float* __restrict__ Vm, const float* __restrict__ Th,
                          float* __restrict__ out)
{
    const int p = blockIdx.x * blockDim.x + threadIdx.x;
    if (p >= EPF) return;
    const int i = pf_i[p], j = pf_j[p];
    const float r  = pf_attr[(size_t)p * 2 + 0];
    const float xr = pf_attr[(size_t)p * 2 + 1];
    const float denom = r * r + xr * xr;
    const float G =  r / denom;
    const float B = -xr / denom;
    const float delta = Th[j] - Th[i];
    float sd, cd;
    sincosf(delta, &sd, &cd);
    const float vv = Vm[i] * Vm[j];
    atomicAdd(&out[(size_t)i * 4 + 2], vv * (G * cd + B * sd));
    atomicAdd(&out[(size_t)i * 4 + 3], vv * (G * sd - B * cd));
}

// ---------------------------------------------------------------------------
// Host launcher
// ---------------------------------------------------------------------------
extern "C" void kernel_launch(void* const* d_in, const int* in_sizes, int n_in,
                              void* d_out, int out_size, void* d_ws, size_t ws_size,
                              hipStream_t stream)
{
    const float* x      = (const float*)d_in[0];
    const float* ea     = (const float*)d_in[1];
    const float* Wq0    = (const float*)d_in[2];
    const float* bq0    = (const float*)d_in[3];
    const float* Wk0    = (const float*)d_in[4];
    const float* bk0    = (const float*)d_in[5];
    const float* Wv0    = (const float*)d_in[6];
    const float* bv0    = (const float*)d_in[7];
    const float* We0    = (const float*)d_in[8];
    const float* Ws0    = (const float*)d_in[9];
    const float* bs0    = (const float*)d_in[10];
    const float* Wq1    = (const float*)d_in[11];
    const float* bq1    = (const float*)d_in[12];
    const float* Wk1    = (const float*)d_in[13];
    const float* bk1    = (const float*)d_in[14];
    const float* Wv1    = (const float*)d_in[15];
    const float* bv1    = (const float*)d_in[16];
    const float* We1    = (const float*)d_in[17];
    const float* Ws1    = (const float*)d_in[18];
    const float* bs1    = (const float*)d_in[19];
    const float* linW   = (const float*)d_in[20];
    const float* pfa    = (const float*)d_in[21];
    const int*   ei     = (const int*)d_in[22];
    const int*   pf_i   = (const int*)d_in[23];
    const int*   pf_j   = (const int*)d_in[24];
    float* out = (float*)d_out;

    // ---- workspace partition (floats) ----
    const size_t ETNC = (size_t)ET * NN * CC;      // 14,680,064
    float* QB    = (float*)d_ws;
    float* KB    = QB    + ETNC;
    float* VB    = KB    + ETNC;
    float* MSG   = VB    + ETNC;
    float* Mseg  = MSG   + ETNC;                   // ET*NN
    float* Sseg  = Mseg  + (size_t)ET * NN;        // ET*NN
    float* LOGIT = Sseg  + (size_t)ET * NN;        // ET*EE
    float* H     = LOGIT + (size_t)ET * EE;        // NT*NN*CC
    float* Vm    = H     + (size_t)NT * NN * CC;   // NT*NN
    float* Th    = Vm    + (size_t)NT * NN;        // NT*NN

    const int GEMM_BLOCKS  = (ET * (NN / 16)) / 8;               // 1792
    const int EDGE_WAVE_BL = (ET * EE) / 8;                      // 114688
    const int EDGE_THR_BL  = (ET * EE) / 256;                    // 3584
    const int SEG_BL       = (ET * NN + 255) / 256;
    const int RED_BL       = (NT * NN * CC) / 256;
    const int NODE_BL      = (NT * NN) / 256;
    const int PF_BL        = EPF / 256;

    // ================= layer 1 (Cin = 11, padded to 12) =================
    proj_gemm_wmma<12, CIN><<<GEMM_BLOCKS, 256, 0, stream>>>(x, Wq0, bq0, QB,  0);
    proj_gemm_wmma<12, CIN><<<GEMM_BLOCKS, 256, 0, stream>>>(x, Wk0, bk0, KB,  1);
    proj_gemm_wmma<12, CIN><<<GEMM_BLOCKS, 256, 0, stream>>>(x, Wv0, bv0, VB,  1);
    proj_gemm_wmma<12, CIN><<<GEMM_BLOCKS, 256, 0, stream>>>(x, Ws0, bs0, MSG, 0);
    init_seg<<<SEG_BL, 256, 0, stream>>>(Mseg, Sseg, ET * NN);
    attn_logit  <<<EDGE_WAVE_BL, 256, 0, stream>>>(QB, KB, ea, We0, ei, LOGIT, Mseg);
    attn_expsum <<<EDGE_THR_BL,  256, 0, stream>>>(ei, LOGIT, Mseg, Sseg);
    attn_scatter<<<EDGE_WAVE_BL, 256, 0, stream>>>(VB, ea, We0, ei, LOGIT, Sseg, MSG);
    reduce_relu <<<RED_BL,       256, 0, stream>>>(MSG, H);

    // ================= layer 2 (Cin = 64) =================
    proj_gemm_wmma<CC, CC><<<GEMM_BLOCKS, 256, 0, stream>>>(H, Wq1, bq1, QB,  0);
    proj_gemm_wmma<CC, CC><<<GEMM_BLOCKS, 256, 0, stream>>>(H, Wk1, bk1, KB,  1);
    proj_gemm_wmma<CC, CC><<<GEMM_BLOCKS, 256, 0, stream>>>(H, Wv1, bv1, VB,  1);
    proj_gemm_wmma<CC, CC><<<GEMM_BLOCKS, 256, 0, stream>>>(H, Ws1, bs1, MSG, 0);
    init_seg<<<SEG_BL, 256, 0, stream>>>(Mseg, Sseg, ET * NN);
    attn_logit  <<<EDGE_WAVE_BL, 256, 0, stream>>>(QB, KB, ea, We1, ei, LOGIT, Mseg);
    attn_expsum <<<EDGE_THR_BL,  256, 0, stream>>>(ei, LOGIT, Mseg, Sseg);
    attn_scatter<<<EDGE_WAVE_BL, 256, 0, stream>>>(VB, ea, We1, ei, LOGIT, Sseg, MSG);
    reduce_relu <<<RED_BL,       256, 0, stream>>>(MSG, H);

    // ================= readout + power flow =================
    final_linear<<<NODE_BL, 256, 0, stream>>>(H, linW, out, Vm, Th);
    powerflow   <<<PF_BL,   256, 0, stream>>>(pfa, pf_i, pf_j, Vm, Th, out);
}